// MyRNNCell_80221399154866
// MI455X (gfx1250) — compile-verified
//
#include <hip/hip_runtime.h>

// Reference output y = concat([x[0], RK4(z)]).  The U@z / W@x[1:] complex
// matvecs in the reference are dead code (not consumed by the return value),
// so the optimal MI455X kernel skips them entirely: that avoids 268 MB of
// HBM traffic (~11.5 us at 23.3 TB/s) for zero output effect.
//
// Live work: 4095 independent lanes, each running 255 fixed-step RK4 steps of
//   dz/dt = (1 - |z|^2) * z
// in fp32 (complex64 == interleaved float pairs).  conj(z)*z has an exactly
// zero imaginary part in float arithmetic, so each f() is a real scalar times
// a complex value — 2 independent FMA chains per lane (re/im), which the
// gfx1250 backend can dual-issue (VOPD) and contract to v_fma_f32.

namespace {
constexpr int kN     = 4096;       // in_sz
constexpr int kSz    = kN - 1;     // 4095 oscillators
constexpr int kSteps = 255;        // int(sqrt(4096))*4 - 1 solution intervals
}

__global__ __launch_bounds__(256) void hopf_rk4_kernel(
    const float*  __restrict__ x_re,   // [4096]
    const float*  __restrict__ x_im,   // [4096]
    const float2* __restrict__ z,      // [4095] complex64 as (re, im) pairs
    float2*       __restrict__ y)      // [4096] complex64 as (re, im) pairs
{
    const int t = blockIdx.x * blockDim.x + threadIdx.x;
    if (t >= kSz) return;

    if (t == 0) {
        // y[0] = x[0] (complex)
        y[0] = make_float2(x_re[0], x_im[0]);
    }

    // Warm the destination cacheline before the long dependent compute chain
    // (emits gfx1250 global_prefetch_b8; the final store then hits in-cache).
    __builtin_prefetch((const void*)(y + t + 1), 0, 1);

    float2 v  = z[t];                  // coalesced global_load_b64
    float  zr = v.x;
    float  zi = v.y;

    // dt computed exactly as the reference: float32(64.0 / 255).
    const float dt  = 64.0f / (float)kSteps;
    const float hdt = 0.5f * dt;
    const float dt6 = dt / 6.0f;

    #pragma unroll 1
    for (int s = 0; s < kSteps; ++s) {
        // k1 = f(z)
        float s1  = 1.0f - (zr * zr + zi * zi);
        float k1r = s1 * zr, k1i = s1 * zi;

        // k2 = f(z + 0.5*dt*k1)
        float ar  = zr + hdt * k1r, ai = zi + hdt * k1i;
        float s2  = 1.0f - (ar * ar + ai * ai);
        float k2r = s2 * ar, k2i = s2 * ai;

        // k3 = f(z + 0.5*dt*k2)
        float br  = zr + hdt * k2r, bi = zi + hdt * k2i;
        float s3  = 1.0f - (br * br + bi * bi);
        float k3r = s3 * br, k3i = s3 * bi;

        // k4 = f(z + dt*k3)
        float cr  = zr + dt * k3r, ci = zi + dt * k3i;
        float s4  = 1.0f - (cr * cr + ci * ci);
        float k4r = s4 * cr, k4i = s4 * ci;

        // z += (dt/6) * (k1 + 2*k2 + 2*k3 + k4)   (reference association order)
        zr = zr + dt6 * (((k1r + 2.0f * k2r) + 2.0f * k3r) + k4r);
        zi = zi + dt6 * (((k1i + 2.0f * k2i) + 2.0f * k3i) + k4i);
    }

    y[t + 1] = make_float2(zr, zi);    // coalesced global_store_b64
}

extern "C" void kernel_launch(void* const* d_in, const int* in_sizes, int n_in,
                              void* d_out, int out_size, void* d_ws, size_t ws_size,
                              hipStream_t stream) {
    (void)in_sizes; (void)n_in; (void)out_size; (void)d_ws; (void)ws_size;

    // setup_inputs() order: x_re (f32), x_im (f32), U (c64, dead), W (c64, dead), z (c64)
    const float*  x_re = (const float*)d_in[0];
    const float*  x_im = (const float*)d_in[1];
    const float2* zptr = (const float2*)d_in[4];   // complex64 -> interleaved floats
    float2*       yptr = (float2*)d_out;           // complex64 -> interleaved floats

    const int threads = 256;                        // 8 wave32 waves per block
    const int blocks  = (kSz + threads - 1) / threads;  // 16 blocks
    hipLaunchKernelGGL(hopf_rk4_kernel, dim3(blocks), dim3(threads), 0, stream,
                       x_re, x_im, zptr, yptr);
}